// HeteroGNN_36172214567613
// MI455X (gfx1250) — compile-verified
//
#include <hip/hip_runtime.h>

typedef float v2f __attribute__((ext_vector_type(2)));
typedef float v8f __attribute__((ext_vector_type(8)));

#define DIMD 128
#define DIMH 64

// ---------------------------------------------------------------- zero
__global__ void zero_kernel(float* __restrict__ p, long long n) {
  long long i = (long long)blockIdx.x * blockDim.x + threadIdx.x;
  long long stride = (long long)gridDim.x * blockDim.x;
  for (; i < n; i += stride) p[i] = 0.0f;
}

// ---------------------------------------------------------------- scatter-add
// One wave per edge: 32 lanes x float4 = 128 floats. L2-resident atomics.
__global__ void scatter_accum_kernel(const float* __restrict__ x_src,
                                     const int* __restrict__ src,
                                     const int* __restrict__ dst,
                                     float* __restrict__ agg,
                                     float* __restrict__ cnt, int E) {
  int wave = (int)((blockIdx.x * blockDim.x + threadIdx.x) >> 5);
  int lane = threadIdx.x & 31;
  if (wave >= E) return;
  int s = src[wave];
  int d = dst[wave];
  const float4* xs = (const float4*)(x_src + (long long)s * DIMD);
  float4 v = xs[lane];
  float* a = agg + (long long)d * DIMD + lane * 4;
  atomicAdd(a + 0, v.x);
  atomicAdd(a + 1, v.y);
  atomicAdd(a + 2, v.z);
  atomicAdd(a + 3, v.w);
  if (lane == 0) atomicAdd(cnt + d, 1.0f);
}

// ---------------------------------------------------------------- fused SAGE GEMM
// out[r,:] = relu((agg[r,:]/max(cnt[r],1)) @ Wl + bias + xdst[r,:] @ Wr).
// One wave computes one 16-row tile with V_WMMA_F32_16X16X4_F32.
// Weights are staged in LDS pre-swizzled to K-pair-interleaved layout so each
// lane's B fragment is a single contiguous ds_load_b64:
//   sW[( (k>>1)*64 + n )*2 + (k&1)] = W[k*64 + n]
__launch_bounds__(256)
__global__ void sage_wmma_kernel(const float* __restrict__ agg,
                                 const float* __restrict__ cnt,
                                 const float* __restrict__ xdst,
                                 const float* __restrict__ Wl,    // [128][64]
                                 const float* __restrict__ Wr,    // [128][64]
                                 const float* __restrict__ bias,  // [64]
                                 float* __restrict__ out,         // [n][64]
                                 int n_tiles, int nrows) {
  __shared__ v2f sWl[(DIMD / 2) * DIMH];  // 32 KB, K-pair interleaved
  __shared__ v2f sWr[(DIMD / 2) * DIMH];  // 32 KB
  {
    float* fl = (float*)sWl;
    float* fr = (float*)sWr;
    for (int i = threadIdx.x; i < DIMD * DIMH; i += blockDim.x) {
      int k = i >> 6;          // row in [0,128)
      int n = i & 63;          // col in [0,64)
      int idx = (((k >> 1) * DIMH + n) << 1) + (k & 1);
      fl[idx] = Wl[i];
      fr[idx] = Wr[i];
    }
  }
  __syncthreads();

  const int wave = threadIdx.x >> 5;
  const int lane = threadIdx.x & 31;
  const int tile = blockIdx.x * (blockDim.x >> 5) + wave;
  if (tile >= n_tiles) return;  // wave-uniform: EXEC stays all-ones for WMMA

  const int n = lane & 15;         // B/C/D column within 16-wide tile
  const int half = lane >> 4;      // 0/1: which K pair of the 4-block
  int arow = tile * 16 + (lane & 15);   // A-fragment row for this lane
  if (arow >= nrows) arow = nrows - 1;  // clamp (keeps EXEC full)

  const v2f* mrow = (const v2f*)(agg + (long long)arow * DIMD);
  const v2f* xrow = (const v2f*)(xdst + (long long)arow * DIMD);
  const float s = 1.0f / fmaxf(cnt[arow], 1.0f);  // fused mean scaling

  // Init accumulators with bias broadcast: C[v] -> (M = v + 8*(lane/16), N)
  v8f acc[4];
#pragma unroll
  for (int nt = 0; nt < 4; ++nt) {
    float b = bias[nt * 16 + n];
#pragma unroll
    for (int v = 0; v < 8; ++v) acc[nt][v] = b;
  }

  for (int kb = 0; kb < DIMD / 4; ++kb) {
    const int kp = (kb << 1) + half;  // K-pair index: k = 2*kp, loads k,k+1
    v2f am = mrow[kp];                // A: aggregated neighbors
    am[0] *= s;
    am[1] *= s;                       // -> mean on the fly
    v2f ax = xrow[kp];                // A: self features
    const v2f* wl0 = sWl + kp * DIMH;
    const v2f* wr0 = sWr + kp * DIMH;
#pragma unroll
    for (int nt = 0; nt < 4; ++nt) {
      v2f bl = wl0[nt * 16 + n];      // single ds_load_b64, adjacent VGPRs
      v2f br = wr0[nt * 16 + n];
      acc[nt] = __builtin_amdgcn_wmma_f32_16x16x4_f32(
          false, am, false, bl, (short)0, acc[nt], false, false);
      acc[nt] = __builtin_amdgcn_wmma_f32_16x16x4_f32(
          false, ax, false, br, (short)0, acc[nt], false, false);
    }
  }

  // Epilogue: relu + store. vgpr v -> row tile*16 + v + 8*(lane/16), col nt*16+n
  const int row_base = tile * 16 + (half << 3);
#pragma unroll
  for (int nt = 0; nt < 4; ++nt) {
#pragma unroll
    for (int v = 0; v < 8; ++v) {
      int row = row_base + v;
      if (row < nrows) {
        float r = acc[nt][v];
        out[(long long)row * DIMH + nt * 16 + n] = r > 0.0f ? r : 0.0f;
      }
    }
  }
}

// ---------------------------------------------------------------- launcher
extern "C" void kernel_launch(void* const* d_in, const int* in_sizes, int n_in,
                              void* d_out, int out_size, void* d_ws,
                              size_t ws_size, hipStream_t stream) {
  (void)n_in; (void)out_size; (void)ws_size;
  const float* x_user = (const float*)d_in[0];
  const float* x_res  = (const float*)d_in[1];
  const int* ua_src   = (const int*)d_in[2];
  const int* ua_dst   = (const int*)d_in[3];
  const int* ru_src   = (const int*)d_in[4];
  const int* ru_dst   = (const int*)d_in[5];
  const float* W_l_ua = (const float*)d_in[6];
  const float* b_ua   = (const float*)d_in[7];
  const float* W_r_ua = (const float*)d_in[8];
  const float* W_l_ru = (const float*)d_in[9];
  const float* b_ru   = (const float*)d_in[10];
  const float* W_r_ru = (const float*)d_in[11];

  const int NU  = in_sizes[0] / DIMD;  // 100000
  const int NR  = in_sizes[1] / DIMD;  // 50000
  const int Eua = in_sizes[2];
  const int Eru = in_sizes[4];

  float* out_user = (float*)d_out;                       // [NU][64] first
  float* out_res  = (float*)d_out + (long long)NU * DIMH;

  // workspace layout (contiguous): agg_res | cnt_res | agg_user | cnt_user
  float* agg_res  = (float*)d_ws;
  float* cnt_res  = agg_res + (long long)NR * DIMD;
  float* agg_user = cnt_res + NR;
  float* cnt_user = agg_user + (long long)NU * DIMD;

  const long long zero_total =
      (long long)NR * DIMD + NR + (long long)NU * DIMD + NU;
  zero_kernel<<<4096, 256, 0, stream>>>(agg_res, zero_total);

  // edge scatter: 8 edges (waves) per 256-thread block
  scatter_accum_kernel<<<(Eua + 7) / 8, 256, 0, stream>>>(
      x_user, ua_src, ua_dst, agg_res, cnt_res, Eua);
  scatter_accum_kernel<<<(Eru + 7) / 8, 256, 0, stream>>>(
      x_res, ru_src, ru_dst, agg_user, cnt_user, Eru);

  // fused mean + GEMM + bias + relu (WMMA f32), one 16-row tile per wave
  const int tiles_res  = (NR + 15) / 16;
  const int tiles_user = (NU + 15) / 16;
  sage_wmma_kernel<<<(tiles_res + 7) / 8, 256, 0, stream>>>(
      agg_res, cnt_res, x_res, W_l_ua, W_r_ua, b_ua, out_res, tiles_res, NR);
  sage_wmma_kernel<<<(tiles_user + 7) / 8, 256, 0, stream>>>(
      agg_user, cnt_user, x_user, W_l_ru, W_r_ru, b_ru, out_user, tiles_user,
      NU);
}